// DiffPool_87187836109057
// MI455X (gfx1250) — compile-verified
//
#include <hip/hip_runtime.h>
#include <hip/hip_bf16.h>

typedef __attribute__((ext_vector_type(2))) float v2f;
typedef __attribute__((ext_vector_type(8))) float v8f;

#define N_NODES 100000
#define N_EDGES 600000
#define N_GRAPHS 64
#define HID 128

__device__ __forceinline__ void atomic_add_f32(float* p, float v) {
  __hip_atomic_fetch_add(p, v, __ATOMIC_RELAXED, __HIP_MEMORY_SCOPE_AGENT);
}

// ---------------------------------------------------------------------------
// GEMM: Out[N,128] = A[N,128] @ W[128,128], fp32 WMMA 16x16x4.
// Block = 256 threads = 8 waves; each wave owns a 16-row x 128-col tile.
// W staged in LDS pair-interleaved so each lane's B operand {W[k][c],W[k+1][c]}
// is one aligned ds_load_b64 directly into an even VGPR pair.
// A rows are clamped (never zero-selected): out-of-range D rows only depend on
// their own A row and are never stored, so unconditional b64 loads are safe.
// ---------------------------------------------------------------------------
__global__ __launch_bounds__(256) void gemm_wmma(const float* __restrict__ A,
                                                 const float* __restrict__ W,
                                                 float* __restrict__ Out,
                                                 int nrows) {
  __shared__ float sW[HID * HID];  // 64 KB, layout: [(k>>1)*256 + col*2 + (k&1)]
  for (int i = threadIdx.x; i < (HID / 2) * HID; i += 256) {
    const int p = i >> 7;       // k-pair index 0..63
    const int col = i & 127;
    v2f w;
    w.x = W[(2 * p) * HID + col];
    w.y = W[(2 * p + 1) * HID + col];
    *(v2f*)&sW[i * 2] = w;      // contiguous 8B write per thread
  }
  __syncthreads();

  const int wave  = threadIdx.x >> 5;      // 0..7
  const int lane  = threadIdx.x & 31;
  const int mrow  = lane & 15;             // M (or N) index within tile
  const int khalf = lane >> 4;             // 0: K pair {0,1}; 1: K pair {2,3}
  const int row0  = blockIdx.x * 128 + wave * 16;

  int arow = row0 + mrow;
  if (arow >= nrows) arow = nrows - 1;     // clamp: garbage rows never stored
  const float* arp = A + (size_t)arow * HID + khalf * 2;

  v8f acc[8];
#pragma unroll
  for (int t = 0; t < 8; ++t) acc[t] = (v8f)0.0f;

  for (int k0 = 0; k0 < HID; k0 += 4) {
    const v2f a = *(const v2f*)(arp + k0);                       // global b64
    const float* bp = &sW[(k0 / 2 + khalf) * 256 + mrow * 2];
#pragma unroll
    for (int t = 0; t < 8; ++t) {
      const v2f b = *(const v2f*)(bp + t * 32);                  // ds b64
      acc[t] = __builtin_amdgcn_wmma_f32_16x16x4_f32(
          /*neg_a=*/false, a, /*neg_b=*/false, b,
          /*c_mod=*/(short)0, acc[t], /*reuse_a=*/false, /*reuse_b=*/false);
    }
  }

  // D layout: VGPR j, lanes 0-15 -> M=j ; lanes 16-31 -> M=8+j ; N = lane&15
#pragma unroll
  for (int t = 0; t < 8; ++t) {
#pragma unroll
    for (int j = 0; j < 8; ++j) {
      const int r = row0 + j + khalf * 8;
      if (r < nrows) Out[(size_t)r * HID + t * 16 + mrow] = acc[t][j];
    }
  }
}

// ---------------------------------------------------------------------------
// Edge scatter: one wave per edge; lane handles 4 contiguous channels.
// Gather hits L2 (m is 51MB < 192MB L2); scatter via native fp32 atomics.
// ---------------------------------------------------------------------------
__global__ __launch_bounds__(256) void scatter_add(const long long* __restrict__ src,
                                                   const long long* __restrict__ dst,
                                                   const float* __restrict__ m,
                                                   float* __restrict__ agg,
                                                   int nedges) {
  const long long gid = (long long)blockIdx.x * blockDim.x + threadIdx.x;
  const long long e = gid >> 5;
  if (e >= nedges) return;
  const int c = ((int)gid & 31) * 4;
  const long long s = src[e];
  const long long d = dst[e];
  const float4 v = *(const float4*)(m + s * HID + c);
  float* o = agg + d * HID + c;
  atomic_add_f32(o + 0, v.x);
  atomic_add_f32(o + 1, v.y);
  atomic_add_f32(o + 2, v.z);
  atomic_add_f32(o + 3, v.w);
}

// ---------------------------------------------------------------------------
// h = relu(agg + bias); accumulate per-channel sum / sumsq (block partials).
// Block = 128 threads (one per channel); rows read coalesced.
// ---------------------------------------------------------------------------
__global__ __launch_bounds__(128) void bias_relu_stats(float* __restrict__ h,
                                                       const float* __restrict__ bias,
                                                       float* __restrict__ sumBuf,
                                                       float* __restrict__ sqBuf,
                                                       int nrows) {
  const int c = threadIdx.x;
  const float b = bias[c];
  float ls = 0.0f, lq = 0.0f;
  const int r0 = blockIdx.x * 128;
  int r1 = r0 + 128;
  if (r1 > nrows) r1 = nrows;
  for (int r = r0; r < r1; ++r) {
    float v = h[(size_t)r * HID + c] + b;
    v = v > 0.0f ? v : 0.0f;
    h[(size_t)r * HID + c] = v;
    ls += v;
    lq += v * v;
  }
  atomic_add_f32(&sumBuf[c], ls);
  atomic_add_f32(&sqBuf[c], lq);
}

__global__ __launch_bounds__(128) void bn_finalize(const float* __restrict__ sum,
                                                   const float* __restrict__ sq,
                                                   const float* __restrict__ g,
                                                   const float* __restrict__ bb,
                                                   float* __restrict__ scale,
                                                   float* __restrict__ shift,
                                                   float invn) {
  const int c = threadIdx.x;
  const float mean = sum[c] * invn;
  const float var = sq[c] * invn - mean * mean;
  const float rstd = rsqrtf(var + 1e-5f);
  const float sc = g[c] * rstd;
  scale[c] = sc;
  shift[c] = bb[c] - mean * sc;
}

__global__ __launch_bounds__(256) void bn_apply(float* __restrict__ h,
                                                const float* __restrict__ scale,
                                                const float* __restrict__ shift,
                                                long long n4) {
  const long long i = (long long)blockIdx.x * blockDim.x + threadIdx.x;
  if (i >= n4) return;
  float4 v = ((float4*)h)[i];
  const int c = (int)((i * 4) & (HID - 1));
  v.x = v.x * scale[c + 0] + shift[c + 0];
  v.y = v.y * scale[c + 1] + shift[c + 1];
  v.z = v.z * scale[c + 2] + shift[c + 2];
  v.w = v.w * scale[c + 3] + shift[c + 3];
  ((float4*)h)[i] = v;
}

__global__ __launch_bounds__(256) void zero_kernel(float* __restrict__ p, long long n) {
  for (long long i = (long long)blockIdx.x * blockDim.x + threadIdx.x; i < n;
       i += (long long)gridDim.x * blockDim.x)
    p[i] = 0.0f;
}

// ---------------------------------------------------------------------------
// Segment-sum pooling. batch[] is sorted, so accumulate runs locally and
// flush to global atomics only on graph-id change.
// ---------------------------------------------------------------------------
__global__ __launch_bounds__(128) void pool_kernel(const float* __restrict__ h,
                                                   const long long* __restrict__ batch,
                                                   float* __restrict__ pooled,
                                                   float* __restrict__ counts,
                                                   int nrows) {
  const int c = threadIdx.x;
  const int r0 = blockIdx.x * 128;
  int r1 = r0 + 128;
  if (r1 > nrows) r1 = nrows;
  float acc = 0.0f;
  float cnt = 0.0f;
  int curg = -1;
  for (int r = r0; r < r1; ++r) {
    const int g = (int)batch[r];
    if (g != curg) {
      if (curg >= 0) {
        atomic_add_f32(&pooled[curg * HID + c], acc);
        if (c == 0) atomic_add_f32(&counts[curg], cnt);
      }
      curg = g;
      acc = 0.0f;
      cnt = 0.0f;
    }
    acc += h[(size_t)r * HID + c];
    cnt += 1.0f;
  }
  if (curg >= 0) {
    atomic_add_f32(&pooled[curg * HID + c], acc);
    if (c == 0) atomic_add_f32(&counts[curg], cnt);
  }
}

// ---------------------------------------------------------------------------
// Head: mean-pool normalize, 128->64 linear, 64->4 linear, log-softmax.
// One thread per graph (64 threads, tiny).
// ---------------------------------------------------------------------------
__global__ __launch_bounds__(64) void head_kernel(const float* __restrict__ pooled,
                                                  const float* __restrict__ counts,
                                                  const float* __restrict__ w1,
                                                  const float* __restrict__ b1,
                                                  const float* __restrict__ w2,
                                                  const float* __restrict__ b2,
                                                  float* __restrict__ out) {
  const int g = threadIdx.x;
  const float icnt = 1.0f / fmaxf(counts[g], 1.0f);
  float h1[64];
  for (int j = 0; j < 64; ++j) {
    float a = b1[j];
    for (int c = 0; c < HID; ++c) a += pooled[g * HID + c] * icnt * w1[c * 64 + j];
    h1[j] = a;
  }
  float o[4];
  for (int j = 0; j < 4; ++j) {
    float a = b2[j];
    for (int c = 0; c < 64; ++c) a += h1[c] * w2[c * 4 + j];
    o[j] = a;
  }
  float mx = fmaxf(fmaxf(o[0], o[1]), fmaxf(o[2], o[3]));
  float se = 0.0f;
  for (int j = 0; j < 4; ++j) se += expf(o[j] - mx);
  const float lse = logf(se) + mx;
  for (int j = 0; j < 4; ++j) out[g * 4 + j] = o[j] - lse;
}

// ---------------------------------------------------------------------------
extern "C" void kernel_launch(void* const* d_in, const int* in_sizes, int n_in,
                              void* d_out, int out_size, void* d_ws, size_t ws_size,
                              hipStream_t stream) {
  const float*     x      = (const float*)d_in[0];
  const long long* ei     = (const long long*)d_in[1];  // [2, E] int64
  const long long* batch  = (const long long*)d_in[2];  // [N] int64
  const float*     conv_w = (const float*)d_in[3];      // [6,128,128]
  const float*     conv_b = (const float*)d_in[4];      // [6,128]
  const float*     bn_g   = (const float*)d_in[5];
  const float*     bn_b   = (const float*)d_in[6];
  const float*     lin1_w = (const float*)d_in[7];
  const float*     lin1_b = (const float*)d_in[8];
  const float*     lin2_w = (const float*)d_in[9];
  const float*     lin2_b = (const float*)d_in[10];

  const int N = N_NODES, E = N_EDGES, G = N_GRAPHS;
  const size_t nh = (size_t)N * HID;  // 12.8M floats

  float* ws     = (float*)d_ws;
  float* bufA   = ws;             // ping-pong h/agg
  float* bufB   = ws + nh;        // m = h @ W
  float* bufC   = ws + 2 * nh;    // ping-pong h/agg
  float* stats  = ws + 3 * nh;    // sum[128] | sq[128] | scale[128] | shift[128]
  float* pooled = stats + 512;    // [G,128]
  float* counts = pooled + (size_t)G * HID;  // [G]

  const int rowBlocks   = (N + 127) / 128;               // 782
  const int scatBlocks  = (int)(((long long)E * 32 + 255) / 256);
  const long long n4    = (long long)(nh / 4);
  const int applyBlocks = (int)((n4 + 255) / 256);

  const float* hin   = x;     // layer input
  float*       aggp  = bufC;  // layer agg/output
  float*       spare = bufA;  // becomes next agg

  for (int i = 0; i < 6; ++i) {
    gemm_wmma<<<rowBlocks, 256, 0, stream>>>(hin, conv_w + (size_t)i * HID * HID, bufB, N);
    zero_kernel<<<8192, 256, 0, stream>>>(aggp, (long long)nh);
    zero_kernel<<<1, 256, 0, stream>>>(stats, 256);
    scatter_add<<<scatBlocks, 256, 0, stream>>>(ei, ei + E, bufB, aggp, E);
    bias_relu_stats<<<rowBlocks, 128, 0, stream>>>(aggp, conv_b + i * HID, stats, stats + 128, N);
    bn_finalize<<<1, 128, 0, stream>>>(stats, stats + 128, bn_g + i * HID, bn_b + i * HID,
                                       stats + 256, stats + 384, 1.0f / (float)N);
    bn_apply<<<applyBlocks, 256, 0, stream>>>(aggp, stats + 256, stats + 384, n4);
    // rotate buffers: aggp becomes the next layer's input
    float* next_spare = (hin == x) ? spare : (float*)hin;
    hin  = aggp;
    aggp = next_spare;
  }

  zero_kernel<<<64, 256, 0, stream>>>(pooled, (long long)G * HID + G);
  pool_kernel<<<rowBlocks, 128, 0, stream>>>(hin, batch, pooled, counts, N);
  head_kernel<<<1, 64, 0, stream>>>(pooled, counts, lin1_w, lin1_b, lin2_w, lin2_b,
                                    (float*)d_out);
}